// GCT_IMPUTER_12841952215442
// MI455X (gfx1250) — compile-verified
//
#include <hip/hip_runtime.h>
#include <hip/hip_bf16.h>

// ---------------------------------------------------------------------------
// Types for WMMA fragments (fp32 path: V_WMMA_F32_16X16X4_F32)
// ---------------------------------------------------------------------------
typedef float v2f __attribute__((ext_vector_type(2)));
typedef float v8f __attribute__((ext_vector_type(8)));

#define HEADS 4
#define CDIM  16
#define HC    64
#define RT    4    // row tiles (of 16 rows) per wave in the GEMM

// ---------------------------------------------------------------------------
// Ordered-uint encoding of float for atomic max (monotonic bijection)
// ---------------------------------------------------------------------------
__device__ __forceinline__ unsigned ordf(float x) {
    unsigned u = __float_as_uint(x);
    return (u & 0x80000000u) ? ~u : (u | 0x80000000u);
}
__device__ __forceinline__ float unordf(unsigned o) {
    unsigned u = (o & 0x80000000u) ? (o ^ 0x80000000u) : ~o;
    return __uint_as_float(u);
}

// ---------------------------------------------------------------------------
// GEMM: Y[N x M] = act(X[N x K] @ W^T + bias),  W stored row-major [M x K].
// One wave32 computes a 64x16 tile of Y (4 row-tiles) via fp32 WMMA,
// reusing the B fragment across the 4 row-tiles each K-step.
// Fragment layouts (32-bit):
//   A 16x4 : lane half (lane>=16) selects K-pair; VGPR pair = consecutive K.
//   B 4x16 : lanes 0-15 hold K={0,1}, lanes 16-31 hold K={2,3} (consecutive).
//   C/D    : VGPR r -> M = r + 8*half, N = lane & 15.
// act: 0 = none, 2 = sigmoid
// ---------------------------------------------------------------------------
__global__ void gemm_wmma_f32(const float* __restrict__ X,
                              const float* __restrict__ W,
                              const float* __restrict__ bias,
                              float* __restrict__ Y,
                              int N, int K, int M, int act) {
    const int lane = threadIdx.x & 31;
    const int wave = (blockIdx.x * blockDim.x + threadIdx.x) >> 5;
    const int rowTiles  = (N + 15) >> 4;
    const int colTiles  = M >> 4;
    const int superRows = (rowTiles + RT - 1) / RT;
    if (wave >= superRows * colTiles) return;          // wave-uniform exit
    const int sr      = wave / colTiles;
    const int colTile = wave % colTiles;

    const int n16  = lane & 15;        // N index within tile (for B/C/D)
    const int half = lane >> 4;        // K-pair selector / M-half for C/D

    // B: lane's N column -> weight row (colTile*16 + n16), length K
    const float* __restrict__ wr = W + (size_t)(colTile * 16 + n16) * K;

    // A rows for the 4 row-tiles (clamped so OOB tiles still load safely;
    // EXEC stays all-ones for the WMMAs)
    const float* __restrict__ xr[RT];
#pragma unroll
    for (int i = 0; i < RT; ++i) {
        int arow = (sr * RT + i) * 16 + n16;
        if (arow >= N) arow = N - 1;
        xr[i] = X + (size_t)arow * K;
    }

    v8f acc[RT] = {};
    for (int k0 = 0; k0 < K; k0 += 4) {
        const int kk = k0 + 2 * half;                  // this lane's K pair
        v2f b = *(const v2f*)(wr + kk);                // one b64 load, reused 4x
#pragma unroll
        for (int i = 0; i < RT; ++i) {
            v2f a = *(const v2f*)(xr[i] + kk);         // one b64 load
            acc[i] = __builtin_amdgcn_wmma_f32_16x16x4_f32(
                /*neg_a=*/false, a, /*neg_b=*/false, b,
                /*c_mod=*/(short)0, acc[i],
                /*reuse_a=*/false, /*reuse_b=*/false);
        }
    }

    const float bv = bias[colTile * 16 + n16];
#pragma unroll
    for (int i = 0; i < RT; ++i) {
#pragma unroll
        for (int r = 0; r < 8; ++r) {
            int orow = (sr * RT + i) * 16 + r + 8 * half;   // C/D: M = r + 8*half
            if (orow < N) {
                float v = acc[i][r] + bv;
                if (act == 2) v = 1.0f / (1.0f + __expf(-v));
                Y[(size_t)orow * M + colTile * 16 + n16] = v;
            }
        }
    }
}

// ---------------------------------------------------------------------------
// Pass 1: per-(edge,head) attention logit + segment max via ordered atomicMax
// ---------------------------------------------------------------------------
__global__ void edge_logits(const float* __restrict__ q,
                            const float* __restrict__ k,
                            const int* __restrict__ ei,
                            float* __restrict__ logits,
                            unsigned* __restrict__ mOrd, int E) {
    int t = blockIdx.x * blockDim.x + threadIdx.x;
    if (t >= E * HEADS) return;
    int e = t >> 2, h = t & 3;
    int src = ei[e], dst = ei[E + e];
    const float4* qp = (const float4*)(q + (size_t)dst * HC + h * CDIM);
    const float4* kp = (const float4*)(k + (size_t)src * HC + h * CDIM);
    float s = 0.0f;
#pragma unroll
    for (int i = 0; i < 4; ++i) {
        float4 a = qp[i], b = kp[i];
        s += a.x * b.x + a.y * b.y + a.z * b.z + a.w * b.w;
    }
    s *= 0.25f;                                      // 1/sqrt(C=16)
    logits[(size_t)e * HEADS + h] = s;
    atomicMax(&mOrd[dst * HEADS + h], ordf(s));
}

// ---------------------------------------------------------------------------
// Pass 2: e = exp(logit - max); store back; segment sum via atomicAdd
// ---------------------------------------------------------------------------
__global__ void edge_exp(const int* __restrict__ ei,
                         float* __restrict__ logits,
                         const unsigned* __restrict__ mOrd,
                         float* __restrict__ denom, int E) {
    int t = blockIdx.x * blockDim.x + threadIdx.x;
    if (t >= E * HEADS) return;
    int e = t >> 2, h = t & 3;
    int dst = ei[E + e];
    float m = unordf(mOrd[dst * HEADS + h]);
    float v = __expf(logits[(size_t)e * HEADS + h] - m);
    logits[(size_t)e * HEADS + h] = v;
    atomicAdd(&denom[dst * HEADS + h], v);
}

// ---------------------------------------------------------------------------
// Pass 3: alpha = e/(denom+1e-16); out[dst] += alpha * v[src]
// ---------------------------------------------------------------------------
__global__ void edge_scatter(const float* __restrict__ vmat,
                             const int* __restrict__ ei,
                             const float* __restrict__ logits,
                             const float* __restrict__ denom,
                             float* __restrict__ out, int E) {
    int t = blockIdx.x * blockDim.x + threadIdx.x;
    if (t >= E * HEADS) return;
    int e = t >> 2, h = t & 3;
    int src = ei[e], dst = ei[E + e];
    float alpha = logits[(size_t)e * HEADS + h] /
                  (denom[dst * HEADS + h] + 1e-16f);
    const float4* vp = (const float4*)(vmat + (size_t)src * HC + h * CDIM);
    float* op = out + (size_t)dst * HC + h * CDIM;
#pragma unroll
    for (int i = 0; i < 4; ++i) {
        float4 mv = vp[i];
        atomicAdd(op + 4 * i + 0, mv.x * alpha);
        atomicAdd(op + 4 * i + 1, mv.y * alpha);
        atomicAdd(op + 4 * i + 2, mv.z * alpha);
        atomicAdd(op + 4 * i + 3, mv.w * alpha);
    }
}

// ---------------------------------------------------------------------------
// Elementwise helpers
// ---------------------------------------------------------------------------
__global__ void zero2_u32(unsigned* __restrict__ a, unsigned* __restrict__ b, int n) {
    int t = blockIdx.x * blockDim.x + threadIdx.x;
    if (t < n) { a[t] = 0u; b[t] = 0u; }
}
__global__ void relu_k(float* __restrict__ a, long n) {
    long t = (long)blockIdx.x * blockDim.x + threadIdx.x;
    if (t < n) a[t] = fmaxf(a[t], 0.0f);
}

// ---------------------------------------------------------------------------
// Host-side orchestration
// ---------------------------------------------------------------------------
extern "C" void kernel_launch(void* const* d_in, const int* in_sizes, int n_in,
                              void* d_out, int out_size, void* d_ws, size_t ws_size,
                              hipStream_t stream) {
    const float* x   = (const float*)d_in[0];
    const int*   ei  = (const int*)d_in[1];
    const float* Wq1 = (const float*)d_in[2];  const float* bq1 = (const float*)d_in[3];
    const float* Wk1 = (const float*)d_in[4];  const float* bk1 = (const float*)d_in[5];
    const float* Wv1 = (const float*)d_in[6];  const float* bv1 = (const float*)d_in[7];
    const float* Ws1 = (const float*)d_in[8];  const float* bs1 = (const float*)d_in[9];
    const float* Wq2 = (const float*)d_in[10]; const float* bq2 = (const float*)d_in[11];
    const float* Wk2 = (const float*)d_in[12]; const float* bk2 = (const float*)d_in[13];
    const float* Wv2 = (const float*)d_in[14]; const float* bv2 = (const float*)d_in[15];
    const float* Ws2 = (const float*)d_in[16]; const float* bs2 = (const float*)d_in[17];
    const float* Wo  = (const float*)d_in[18]; const float* bo  = (const float*)d_in[19];

    const int D = 256;
    const int N = in_sizes[0] / D;
    const int E = in_sizes[1] / 2;

    // Carve workspace
    char* p = (char*)d_ws;
    auto carve = [&](size_t bytes) -> void* {
        void* r = (void*)p;
        p += (bytes + 255) & ~(size_t)255;
        return r;
    };
    float*    qb     = (float*)carve((size_t)N * HC * sizeof(float));
    float*    kb     = (float*)carve((size_t)N * HC * sizeof(float));
    float*    vb     = (float*)carve((size_t)N * HC * sizeof(float));
    float*    hA     = (float*)carve((size_t)N * HC * sizeof(float));
    float*    hB     = (float*)carve((size_t)N * HC * sizeof(float));
    float*    logits = (float*)carve((size_t)E * HEADS * sizeof(float));
    unsigned* mOrd   = (unsigned*)carve((size_t)N * HEADS * sizeof(unsigned));
    float*    denom  = (float*)carve((size_t)N * HEADS * sizeof(float));

    const int TB = 256;
    auto gemm = [&](const float* X, const float* W, const float* b, float* Y,
                    int K, int M, int act) {
        int rowTiles  = (N + 15) >> 4;
        int colTiles  = M >> 4;
        int superRows = (rowTiles + RT - 1) / RT;
        long waves    = (long)superRows * colTiles;
        long threads  = waves * 32;
        int  blocks   = (int)((threads + TB - 1) / TB);
        gemm_wmma_f32<<<blocks, TB, 0, stream>>>(X, W, b, Y, N, K, M, act);
    };

    const int nh       = N * HEADS;
    const int nhBlk    = (nh + TB - 1) / TB;
    const int ehBlk    = (E * HEADS + TB - 1) / TB;
    const long nodeEl  = (long)N * HC;
    const int  nodeBlk = (int)((nodeEl + TB - 1) / TB);

    // ---------------- Layer 1 (input x, K = 256) ----------------
    zero2_u32<<<nhBlk, TB, 0, stream>>>(mOrd, (unsigned*)denom, nh);
    gemm(x, Wq1, bq1, qb, D, HC, 0);
    gemm(x, Wk1, bk1, kb, D, HC, 0);
    gemm(x, Wv1, bv1, vb, D, HC, 0);
    gemm(x, Ws1, bs1, hA, D, HC, 0);                 // skip term initializes output
    edge_logits <<<ehBlk, TB, 0, stream>>>(qb, kb, ei, logits, mOrd, E);
    edge_exp    <<<ehBlk, TB, 0, stream>>>(ei, logits, mOrd, denom, E);
    edge_scatter<<<ehBlk, TB, 0, stream>>>(vb, ei, logits, denom, hA, E);
    relu_k      <<<nodeBlk, TB, 0, stream>>>(hA, nodeEl);

    // ---------------- Layer 2 (input hA, K = 64) ----------------
    zero2_u32<<<nhBlk, TB, 0, stream>>>(mOrd, (unsigned*)denom, nh);
    gemm(hA, Wq2, bq2, qb, HC, HC, 0);
    gemm(hA, Wk2, bk2, kb, HC, HC, 0);
    gemm(hA, Wv2, bv2, vb, HC, HC, 0);
    gemm(hA, Ws2, bs2, hB, HC, HC, 0);
    edge_logits <<<ehBlk, TB, 0, stream>>>(qb, kb, ei, logits, mOrd, E);
    edge_exp    <<<ehBlk, TB, 0, stream>>>(ei, logits, mOrd, denom, E);
    edge_scatter<<<ehBlk, TB, 0, stream>>>(vb, ei, logits, denom, hB, E);
    relu_k      <<<nodeBlk, TB, 0, stream>>>(hB, nodeEl);

    // ---------------- Output projection + sigmoid ----------------
    gemm(hB, Wo, bo, (float*)d_out, HC, D, 2);
}